// LiftSplatShootEncoder_82506321756932
// MI455X (gfx1250) — compile-verified
//
#include <hip/hip_runtime.h>
#include <hip/hip_bf16.h>

// ---------------- problem constants ----------------
constexpr int Bb  = 4;
constexpr int Nc  = 6;
constexpr int Dd  = 41;
constexpr int FH  = 16;
constexpr int FW  = 44;
constexpr int Cc  = 64;
constexpr int NX  = 200;
constexpr int NY  = 200;
constexpr int NPRIME = Bb * Nc * Dd * FH * FW;      // 692,736
constexpr int PLANE  = NY * NX;                     // 40,000
constexpr int OUT_N  = Bb * Cc * NY * NX;           // 10,240,000
constexpr int MAT_STRIDE = 80;                      // floats per (b,n) record

typedef float v2f __attribute__((ext_vector_type(2)));
typedef float v8f __attribute__((ext_vector_type(8)));

// ---------------- 3x3 inverse helper ----------------
__device__ inline void inv3(const float* a, float* o) {
    float a00=a[0],a01=a[1],a02=a[2],a10=a[3],a11=a[4],a12=a[5],a20=a[6],a21=a[7],a22=a[8];
    float c00 =   a11*a22 - a12*a21;
    float c01 = -(a10*a22 - a12*a20);
    float c02 =   a10*a21 - a11*a20;
    float id  = 1.0f / (a00*c00 + a01*c01 + a02*c02);
    o[0] =  c00*id;
    o[1] = -(a01*a22 - a02*a21)*id;
    o[2] =  (a01*a12 - a02*a11)*id;
    o[3] =  c01*id;
    o[4] =  (a00*a22 - a02*a20)*id;
    o[5] = -(a00*a12 - a02*a10)*id;
    o[6] =  c02*id;
    o[7] = -(a00*a21 - a01*a20)*id;
    o[8] =  (a00*a11 - a01*a10)*id;
}

// ---------------- per-(b,n) matrix precompute ----------------
// mats record layout (80 floats):
//  [0..8]   inv(post_rots), row-major
//  [9..11]  post_trans
//  [12..43] WMMA A-operand VGPR0 value per lane (a0 table)
//  [44..75] WMMA A-operand VGPR1 value per lane (a1 table)
// A (16x4) = rows 0..2 = [combine | trans], rows 3..15 = 0
// 32-bit A layout: lanes 0-15 hold K=0(a0)/K=1(a1); lanes 16-31 hold K=2(a0)/K=3(a1)
__global__ void lss_prep_kernel(const float* __restrict__ rots,
                                const float* __restrict__ trans,
                                const float* __restrict__ intrins,
                                const float* __restrict__ post_rots,
                                const float* __restrict__ post_trans,
                                float* __restrict__ mats) {
    int bn = blockIdx.x * blockDim.x + threadIdx.x;
    if (bn >= Bb * Nc) return;

    float iPR[9], iI[9], comb[9];
    inv3(post_rots + bn * 9, iPR);
    inv3(intrins   + bn * 9, iI);
    const float* R = rots + bn * 9;
    #pragma unroll
    for (int i = 0; i < 3; ++i)
        #pragma unroll
        for (int j = 0; j < 3; ++j)
            comb[i*3+j] = R[i*3+0]*iI[0*3+j] + R[i*3+1]*iI[1*3+j] + R[i*3+2]*iI[2*3+j];

    float* M = mats + bn * MAT_STRIDE;
    #pragma unroll
    for (int i = 0; i < 9; ++i) M[i] = iPR[i];
    #pragma unroll
    for (int i = 0; i < 3; ++i) M[9+i] = post_trans[bn*3+i];

    // zero A tables then fill rows 0..2
    #pragma unroll
    for (int l = 0; l < 32; ++l) { M[12+l] = 0.0f; M[44+l] = 0.0f; }
    #pragma unroll
    for (int r = 0; r < 3; ++r) {
        M[12 + r]      = comb[r*3+0];      // lane r,     K=0
        M[44 + r]      = comb[r*3+1];      // lane r,     K=1
        M[12 + 16 + r] = comb[r*3+2];      // lane r+16,  K=2
        M[44 + 16 + r] = trans[bn*3+r];    // lane r+16,  K=3
    }
}

// ---------------- geometry + voxel quantize (WMMA) ----------------
// One wave handles 16 consecutive frustum points (camera-uniform group).
// Lanes L and L+16 both compute point (base + L%16); the WMMA computes
// D = A(3x4 camera matrix, padded) x B(4x16 homogeneous points):
// lane j<16 then holds its own point's ego-frame X,Y,Z in D[0..2].
__global__ __launch_bounds__(128) void lss_geom_kernel(const float* __restrict__ mats,
                                                       int* __restrict__ vidx) {
    const int lane  = threadIdx.x & 31;
    const int m     = lane & 15;
    const int gwave = blockIdx.x * 4 + (threadIdx.x >> 5);
    const int point = gwave * 16 + m;

    // decode (b,n,d,h,w)
    int t = point;
    const int w = t % FW; t /= FW;
    const int h = t % FH; t /= FH;
    const int d = t % Dd; t /= Dd;
    const int bn = __builtin_amdgcn_readfirstlane(t);   // uniform across the wave
    const int b  = bn / Nc;

    const float* __restrict__ M = mats + bn * MAT_STRIDE;

    // frustum pixel/depth coordinates (match jnp.linspace)
    const float u   = (float)w * (703.0f / 43.0f);
    const float v   = (float)h * 17.0f;
    const float dep = 2.0f + (float)d;

    // undo post-augmentation: invPostRots @ (p - post_trans)
    const float p0 = u   - M[9];
    const float p1 = v   - M[10];
    const float p2 = dep - M[11];
    const float q0 = M[0]*p0 + M[1]*p1 + M[2]*p2;
    const float q1 = M[3]*p0 + M[4]*p1 + M[5]*p2;
    const float q2 = M[6]*p0 + M[7]*p1 + M[8]*p2;

    // un-perspective: (u*d, v*d, d)
    const float x3 = q0 * q2;
    const float y3 = q1 * q2;
    const float z3 = q2;

    // A operand: precomputed per-lane table (rows 0..2 = [combine|trans])
    v2f Av;
    Av[0] = M[12 + lane];
    Av[1] = M[44 + lane];

    // B operand (4x16): column j = (x,y,z,1) of point j.
    // lanes 0-15: K=0 -> x, K=1 -> y; lanes 16-31: K=2 -> z, K=3 -> 1
    const bool hi = lane >= 16;
    v2f Bv;
    Bv[0] = hi ? z3 : x3;
    Bv[1] = hi ? 1.0f : y3;

    v8f Cv = {};
    // D = A x B + 0 :  v_wmma_f32_16x16x4_f32
    v8f Dv = __builtin_amdgcn_wmma_f32_16x16x4_f32(
        /*neg_a=*/false, Av, /*neg_b=*/false, Bv,
        /*c_mod=*/(short)0, Cv, /*reuse_a=*/false, /*reuse_b=*/false);

    // lane j (j<16): Dv[0]=X, Dv[1]=Y, Dv[2]=Z of point j  (D: VGPR r -> M=r, N=lane)
    const float X = Dv[0];
    const float Y = Dv[1];
    const float Z = Dv[2];

    // quantize: (geom - (BX - DX/2)) / DX, truncate toward zero (matches .astype(int32))
    const int gx = (int)((X + 50.0f) / 0.5f);
    const int gy = (int)((Y + 50.0f) / 0.5f);
    const int gz = (int)((Z + 10.0f) / 20.0f);

    const bool kept = (gx >= 0) & (gx < NX) & (gy >= 0) & (gy < NY) & (gz == 0);
    const int out_base = b * (Cc * PLANE) + gy * NX + gx;   // channel offset added later
    const int val = kept ? out_base : -1;

    if (lane < 16) vidx[point] = val;
}

// ---------------- zero the output grid ----------------
__global__ __launch_bounds__(256) void lss_zero_kernel(float4* __restrict__ out, int n4) {
    int i = blockIdx.x * blockDim.x + threadIdx.x;
    if (i < n4) out[i] = make_float4(0.0f, 0.0f, 0.0f, 0.0f);
}

// ---------------- feature scatter-add ----------------
// One wave per point: lane reads 2 channels (float2 = 256B/wave, fully coalesced),
// then 2 hardware f32 atomics into the BEV grid (41 MB grid lives in the 192 MB L2).
__global__ __launch_bounds__(256) void lss_scatter_kernel(const float* __restrict__ x,
                                                          const int* __restrict__ vidx,
                                                          float* __restrict__ out) {
    const int lane  = threadIdx.x & 31;
    const int point = (blockIdx.x * blockDim.x + threadIdx.x) >> 5;

    const int vi = __builtin_amdgcn_readfirstlane(vidx[point]);
    if (vi < 0) return;   // dropped point: reference adds zeros (no-op)

    const float2 f = ((const float2*)(x + (size_t)point * Cc))[lane];
    float* o = out + vi + (size_t)(2 * lane) * PLANE;
    unsafeAtomicAdd(o,         f.x);   // global_atomic_add_f32
    unsafeAtomicAdd(o + PLANE, f.y);
}

// ---------------- launcher ----------------
extern "C" void kernel_launch(void* const* d_in, const int* in_sizes, int n_in,
                              void* d_out, int out_size, void* d_ws, size_t ws_size,
                              hipStream_t stream) {
    const float* x          = (const float*)d_in[0];
    const float* rots       = (const float*)d_in[1];
    const float* trans      = (const float*)d_in[2];
    const float* intrins    = (const float*)d_in[3];
    const float* post_rots  = (const float*)d_in[4];
    const float* post_trans = (const float*)d_in[5];
    float* out = (float*)d_out;

    // scratch layout: [NPRIME ints: voxel index per point][24 * 80 floats: matrices]
    int*   vidx = (int*)d_ws;
    float* mats = (float*)((char*)d_ws + (size_t)NPRIME * sizeof(int));

    // 1) per-camera matrices (24 of them)
    lss_prep_kernel<<<1, 32, 0, stream>>>(rots, trans, intrins, post_rots, post_trans, mats);

    // 2) geometry via WMMA: NPRIME/16 = 43,296 waves, 4 waves/block
    lss_geom_kernel<<<NPRIME / 16 / 4, 128, 0, stream>>>(mats, vidx);

    // 3) zero output (harness poisons d_out)
    constexpr int N4 = OUT_N / 4;
    lss_zero_kernel<<<(N4 + 255) / 256, 256, 0, stream>>>((float4*)out, N4);

    // 4) scatter-add features: 1 wave per point, 8 waves/block
    lss_scatter_kernel<<<NPRIME / 8, 256, 0, stream>>>(x, vidx, out);
}